// KBRD_84602265797144
// MI455X (gfx1250) — compile-verified
//
#include <hip/hip_runtime.h>
#include <hip/hip_bf16.h>

#define N_ENTITY 64368
#define N_REL    40
#define DIM      128
#define N_BASES  8
#define N_EDGES  500000
#define BATCH    64
#define SEED_LEN 32
#define N_TILES  (N_ENTITY / 16)   // 4023, exact

typedef __attribute__((ext_vector_type(2))) float v2f;
typedef __attribute__((ext_vector_type(8))) float v8f;

// Native device-scope f32 atomic add (lowers to global_atomic_add_f32,
// non-returning since the result is discarded).
__device__ __forceinline__ void atomic_add_f32(float* p, float v) {
    __hip_atomic_fetch_add(p, v, __ATOMIC_RELAXED, __HIP_MEMORY_SCOPE_AGENT);
}

// ---------------------------------------------------------------------------
// Kernel 1: zero the aggregation buffer and degree counters (must run every
// launch: we accumulate with atomics and the harness does not re-poison ws).
// ---------------------------------------------------------------------------
__global__ void kbrd_zero(float* __restrict__ agg, float* __restrict__ deg) {
    const int stride = gridDim.x * blockDim.x;
    const int tid = blockIdx.x * blockDim.x + threadIdx.x;
    const int total = N_ENTITY * DIM;
    for (int i = tid; i < total; i += stride) agg[i] = 0.0f;
    for (int i = tid; i < N_ENTITY; i += stride) deg[i] = 0.0f;
}

// ---------------------------------------------------------------------------
// Kernel 2: per-edge message + scatter.  One wave (32 lanes) per edge; each
// lane owns 4 contiguous floats of the 128-dim message (float4 loads of the
// basis rows), then 4 native global f32 atomics into agg[dst].
// ---------------------------------------------------------------------------
__global__ void kbrd_edge_scatter(const int* __restrict__ esrc,
                                  const int* __restrict__ edst,
                                  const int* __restrict__ etype,
                                  const float* __restrict__ basis,
                                  const float* __restrict__ att,
                                  float* __restrict__ agg,
                                  float* __restrict__ deg) {
    const int lane   = threadIdx.x & 31;
    const int wave   = (blockIdx.x * blockDim.x + threadIdx.x) >> 5;
    const int nwaves = (gridDim.x * blockDim.x) >> 5;
    const int j4 = lane * 4;

    for (int e = wave; e < N_EDGES; e += nwaves) {
        const int src = esrc[e];
        const int dst = edst[e];
        const int t   = etype[e];
        const float* cf = att + t * N_BASES;
        float c[N_BASES];
#pragma unroll
        for (int b = 0; b < N_BASES; ++b) c[b] = cf[b];

        const float* bp = basis + (size_t)src * DIM + j4;
        float mx = 0.f, my = 0.f, mz = 0.f, mw = 0.f;
#pragma unroll
        for (int b = 0; b < N_BASES; ++b) {
            const float4 v = *(const float4*)(bp + (size_t)b * (size_t)N_ENTITY * DIM);
            mx = fmaf(c[b], v.x, mx);
            my = fmaf(c[b], v.y, my);
            mz = fmaf(c[b], v.z, mz);
            mw = fmaf(c[b], v.w, mw);
        }
        float* ap = agg + (size_t)dst * DIM + j4;
        atomic_add_f32(ap + 0, mx);
        atomic_add_f32(ap + 1, my);
        atomic_add_f32(ap + 2, mz);
        atomic_add_f32(ap + 3, mw);
        if (lane == 0) atomic_add_f32(deg + dst, 1.0f);
    }
}

// ---------------------------------------------------------------------------
// Kernel 3: nodes = agg / max(deg,1) + root + bias   (in place on agg)
// ---------------------------------------------------------------------------
__global__ void kbrd_finalize(float* __restrict__ nodes,
                              const float* __restrict__ deg,
                              const float* __restrict__ root,
                              const float* __restrict__ bias) {
    const int stride = gridDim.x * blockDim.x;
    const int total = N_ENTITY * DIM;
    for (int idx = blockIdx.x * blockDim.x + threadIdx.x; idx < total; idx += stride) {
        const int i = idx >> 7;        // / DIM
        const int j = idx & (DIM - 1); // % DIM
        float d = deg[i];
        d = d < 1.0f ? 1.0f : d;
        nodes[idx] = nodes[idx] / d + root[idx] + bias[j];
    }
}

// ---------------------------------------------------------------------------
// Kernel 4: attention pooling.  One 128-thread block per batch element.
// H = nodes[seed_ids[b]]; e = tanh(H @ A) @ b; attn = softmax(e); u = attn@H
// ---------------------------------------------------------------------------
__global__ void kbrd_attn_pool(const float* __restrict__ nodes,
                               const int* __restrict__ seed_ids,
                               const float* __restrict__ A,
                               const float* __restrict__ Bv,
                               float* __restrict__ U) {
    __shared__ float H[SEED_LEN][DIM];
    __shared__ float red[DIM];
    __shared__ float ebuf[SEED_LEN];
    __shared__ float attn[SEED_LEN];

    const int b = blockIdx.x;
    const int d = threadIdx.x;  // 0..127

    for (int s = 0; s < SEED_LEN; ++s) {
        const int id = seed_ids[b * SEED_LEN + s];
        H[s][d] = nodes[(size_t)id * DIM + d];
    }
    __syncthreads();

    const float bw = Bv[d];
    for (int s = 0; s < SEED_LEN; ++s) {
        float acc = 0.0f;
#pragma unroll 8
        for (int k = 0; k < DIM; ++k)
            acc = fmaf(H[s][k], A[k * DIM + d], acc);
        red[d] = tanhf(acc) * bw;
        __syncthreads();
        for (int off = 64; off >= 1; off >>= 1) {
            if (d < off) red[d] += red[d + off];
            __syncthreads();
        }
        if (d == 0) ebuf[s] = red[0];
        __syncthreads();
    }

    if (d == 0) {
        float mx = -1e30f;
        for (int s = 0; s < SEED_LEN; ++s) mx = fmaxf(mx, ebuf[s]);
        float sum = 0.0f;
        for (int s = 0; s < SEED_LEN; ++s) { attn[s] = __expf(ebuf[s] - mx); sum += attn[s]; }
        const float inv = 1.0f / sum;
        for (int s = 0; s < SEED_LEN; ++s) attn[s] *= inv;
    }
    __syncthreads();

    float u = 0.0f;
    for (int s = 0; s < SEED_LEN; ++s) u = fmaf(attn[s], H[s][d], u);
    U[b * DIM + d] = u;
}

// ---------------------------------------------------------------------------
// Kernel 5: scores = U[64,128] @ nodes.T[128,N_ENTITY] + out_bias
// V_WMMA_F32_16X16X4_F32: each wave owns one 16-entity column tile and all
// four 16-row tiles of U; K=128 in steps of 4 -> 128 WMMAs per wave.
//
// A 16x4 f32 layout (ISA 7.12.2): lane M = l&15; VGPR0 = A[M][K0 + (l<16?0:2)],
// VGPR1 = A[M][K0 + (l<16?1:3)].  B 4x16 mirrored: lane N = l&15, same K split.
// C/D: VGPR v, lane l -> row v + (l<16?0:8), col l&15.
// ---------------------------------------------------------------------------
__global__ void kbrd_scores_wmma(const float* __restrict__ nodes,
                                 const float* __restrict__ U,
                                 const float* __restrict__ out_bias,
                                 float* __restrict__ out) {
    const int lane  = threadIdx.x & 31;
    const int ntile = blockIdx.x * 8 + (threadIdx.x >> 5);
    if (ntile >= N_TILES) return;

    const int nbase = ntile * 16;
    const int nm    = lane & 15;          // N col (B/D) and M row (A)
    const int khalf = (lane >> 4) << 1;   // 0 or 2

    v8f c0 = {}, c1 = {}, c2 = {}, c3 = {};
    const float* bp = nodes + (size_t)(nbase + nm) * DIM;   // B column = nodes row

#pragma unroll 4
    for (int kk = 0; kk < DIM; kk += 4) {
        v2f bf;
        bf.x = bp[kk + khalf];
        bf.y = bp[kk + khalf + 1];

        const float* up = U + nm * DIM + kk + khalf;
        v2f a0; a0.x = up[0 * 16 * DIM]; a0.y = up[0 * 16 * DIM + 1];
        v2f a1; a1.x = up[1 * 16 * DIM]; a1.y = up[1 * 16 * DIM + 1];
        v2f a2; a2.x = up[2 * 16 * DIM]; a2.y = up[2 * 16 * DIM + 1];
        v2f a3; a3.x = up[3 * 16 * DIM]; a3.y = up[3 * 16 * DIM + 1];

        c0 = __builtin_amdgcn_wmma_f32_16x16x4_f32(false, a0, false, bf, (short)0, c0, false, false);
        c1 = __builtin_amdgcn_wmma_f32_16x16x4_f32(false, a1, false, bf, (short)0, c1, false, false);
        c2 = __builtin_amdgcn_wmma_f32_16x16x4_f32(false, a2, false, bf, (short)0, c2, false, false);
        c3 = __builtin_amdgcn_wmma_f32_16x16x4_f32(false, a3, false, bf, (short)0, c3, false, false);
    }

    const float bias = out_bias[nbase + nm];
    const int rbase = (lane < 16) ? 0 : 8;
    float* op = out + nbase + nm;
#pragma unroll
    for (int v = 0; v < 8; ++v) {
        const int m = v + rbase;
        op[(size_t)(0 * 16 + m) * N_ENTITY] = c0[v] + bias;
        op[(size_t)(1 * 16 + m) * N_ENTITY] = c1[v] + bias;
        op[(size_t)(2 * 16 + m) * N_ENTITY] = c2[v] + bias;
        op[(size_t)(3 * 16 + m) * N_ENTITY] = c3[v] + bias;
    }
}

// ---------------------------------------------------------------------------
extern "C" void kernel_launch(void* const* d_in, const int* in_sizes, int n_in,
                              void* d_out, int out_size, void* d_ws, size_t ws_size,
                              hipStream_t stream) {
    const int*   seed_ids  = (const int*)d_in[0];
    const int*   edge_src  = (const int*)d_in[1];
    const int*   edge_dst  = (const int*)d_in[2];
    const int*   edge_type = (const int*)d_in[3];
    const float* basis     = (const float*)d_in[4];
    const float* att       = (const float*)d_in[5];
    const float* root      = (const float*)d_in[6];
    const float* rgcn_bias = (const float*)d_in[7];
    const float* attn_a    = (const float*)d_in[8];
    const float* attn_b    = (const float*)d_in[9];
    const float* out_bias  = (const float*)d_in[10];
    float* out = (float*)d_out;

    float* agg = (float*)d_ws;                          // N_ENTITY*DIM
    float* deg = agg + (size_t)N_ENTITY * DIM;          // N_ENTITY
    float* U   = deg + N_ENTITY;                        // BATCH*DIM
    (void)in_sizes; (void)n_in; (void)out_size; (void)ws_size;

    kbrd_zero<<<1024, 256, 0, stream>>>(agg, deg);
    kbrd_edge_scatter<<<2048, 256, 0, stream>>>(edge_src, edge_dst, edge_type,
                                                basis, att, agg, deg);
    kbrd_finalize<<<1024, 256, 0, stream>>>(agg, deg, root, rgcn_bias);
    kbrd_attn_pool<<<BATCH, DIM, 0, stream>>>(agg, seed_ids, attn_a, attn_b, U);
    kbrd_scores_wmma<<<(N_TILES + 7) / 8, 256, 0, stream>>>(agg, U, out_bias, out);
}